// VectorQuantization_21517786153013
// MI455X (gfx1250) — compile-verified
//
#include <hip/hip_runtime.h>
#include <hip/hip_bf16.h>

typedef __attribute__((ext_vector_type(2))) float v2f;
typedef __attribute__((ext_vector_type(8))) float v8f;

#define CDIM 64
#define KCODES 512
#define HW 4096          // H*W
#define NTOT 262144      // B*H*W
#define ROWSTRIDE 68     // padded LDS row stride (bank-conflict avoidance)

// ---------------------------------------------------------------------------
// ws layout (floats):
//   [0]            diff accumulator
//   [1..512]       emb_norm[k] = sum_c e[c,k]^2
//   [513..1024]    counts[k]
//   [1025..33792]  embsum[c*512+k]
//   [33793..34304] cs_scaled[k]
// ---------------------------------------------------------------------------

__global__ void vq_init_kernel(const float* __restrict__ emb, float* __restrict__ ws) {
    int g = blockIdx.x * blockDim.x + threadIdx.x;   // 0..32767
    if (g == 0) ws[0] = 0.0f;
    if (g < KCODES) {
        float s = 0.0f;
        #pragma unroll 8
        for (int c = 0; c < CDIM; ++c) {
            float v = emb[c * KCODES + g];
            s += v * v;
        }
        ws[1 + g] = s;         // emb_norm
        ws[513 + g] = 0.0f;    // counts
    }
    ws[1025 + g] = 0.0f;       // embsum (32768 elems)
}

// ---------------------------------------------------------------------------
// Main fused kernel: distances via V_WMMA_F32_16X16X4_F32, argmin, gather,
// quantize write-out (NCHW), diff partial, counts / embedding_sum atomics.
// Block = 256 threads = 8 waves; wave w handles one 16-row tile of flat.
// Dynamic LDS = full embedding (128 KB).
// ---------------------------------------------------------------------------
__global__ void __launch_bounds__(256)
vq_main_kernel(const float* __restrict__ input,      // [B,C,H,W]
               const float* __restrict__ embedding,  // [C,K]
               const float* __restrict__ embNorm,    // [K]
               float* __restrict__ counts,           // [K] accum
               float* __restrict__ embsum,           // [C*K] accum
               float* __restrict__ wsDiff,           // [1] accum
               float* __restrict__ outQ,             // [B,C,H,W]
               float* __restrict__ outInd)           // [B*H*W] (as float)
{
    extern __shared__ float semb[];                  // 64*512 = 128KB
    __shared__ float sflat[8 * 16 * ROWSTRIDE];      // per-wave 16x64 tiles, padded
    __shared__ int   sIdx[8 * 16];                   // per-wave argmin results

    const int tid  = threadIdx.x;
    const int w    = tid >> 5;
    const int lane = tid & 31;
    const int colLane = lane & 15;
    const int half    = lane >> 4;                   // 0: lanes 0-15, 1: lanes 16-31

    // ---- stage embedding into LDS (cooperative, coalesced) ----
    for (int i = tid; i < CDIM * KCODES; i += 256) semb[i] = embedding[i];

    // ---- stage this wave's 16x64 input tile into LDS (transposed gather) ----
    const int tile = blockIdx.x * 8 + w;             // 0..16383
    const int n0   = tile << 4;                      // first flat row
    const int bImg = n0 >> 12;                       // n0 / HW  (tiles never straddle images)
    const int hw0  = n0 & (HW - 1);
    const float* inBase = input + (size_t)bImg * CDIM * HW;
    float* myFlat = sflat + w * (16 * ROWSTRIDE);

    #pragma unroll
    for (int i = 0; i < 32; ++i) {
        int e = i * 32 + lane;                       // 0..1023
        int c = e >> 4;                              // lanes 0-15: c=2i, 16-31: c=2i+1
        int r = e & 15;                              // 16 consecutive floats per half-wave
        myFlat[r * ROWSTRIDE + c] = inBase[c * HW + hw0 + r];
    }
    __syncthreads();

    // ---- GEMM + fused argmin:  argmin_k ( ||e_k||^2 - 2 * x . e_k ) ----
    float bestD[8];
    int   bestK[8];
    #pragma unroll
    for (int j = 0; j < 8; ++j) { bestD[j] = 3.4e38f; bestK[j] = 0; }

    for (int kt = 0; kt < 32; ++kt) {
        const int k0 = kt << 4;
        v8f acc = {};
        #pragma unroll
        for (int c0 = 0; c0 < CDIM; c0 += 4) {
            // A fragment: 16x4 f32, row = colLane, K-pair selected by lane half
            int ca = c0 + 2 * half;
            v2f a, b;
            a.x = myFlat[colLane * ROWSTRIDE + ca];
            a.y = myFlat[colLane * ROWSTRIDE + ca + 1];
            // B fragment: 4x16 f32, col = colLane, K-pair selected by lane half
            b.x = semb[(ca)     * KCODES + k0 + colLane];
            b.y = semb[(ca + 1) * KCODES + k0 + colLane];
            acc = __builtin_amdgcn_wmma_f32_16x16x4_f32(
                false, a, false, b, (short)0, acc, false, false);
        }
        // dist = ||e||^2 - 2*score ; lane covers code k0+colLane, rows j (+8 for half 1)
        const float nrm = embNorm[k0 + colLane];
        const int   kk  = k0 + colLane;
        #pragma unroll
        for (int j = 0; j < 8; ++j) {
            float d = nrm - 2.0f * acc[j];
            if (d < bestD[j]) { bestD[j] = d; bestK[j] = kk; }   // kt ascending -> first min kept
        }
    }

    // ---- cross-lane argmin reduce within each 16-lane half (wave32 shuffles) ----
    #pragma unroll
    for (int j = 0; j < 8; ++j) {
        #pragma unroll
        for (int s = 1; s < 16; s <<= 1) {
            float od = __shfl_xor(bestD[j], s, 32);
            int   ok = __shfl_xor(bestK[j], s, 32);
            if (od < bestD[j] || (od == bestD[j] && ok < bestK[j])) {
                bestD[j] = od; bestK[j] = ok;
            }
        }
    }
    if (colLane == 0) {                              // lanes 0 (rows 0-7) and 16 (rows 8-15)
        #pragma unroll
        for (int j = 0; j < 8; ++j) sIdx[w * 16 + half * 8 + j] = bestK[j];
    }
    __syncthreads();

    // ---- gather codes, write quantized NCHW output, diff + EMA statistics ----
    float dAcc = 0.0f;
    #pragma unroll
    for (int i = 0; i < 32; ++i) {
        int e = i * 32 + lane;
        int c = e >> 4;
        int r = e & 15;
        int kq = sIdx[w * 16 + r];
        float q = semb[c * KCODES + kq];
        float x = myFlat[r * ROWSTRIDE + c];
        outQ[(size_t)(bImg * CDIM + c) * HW + hw0 + r] = q;   // straight-through == q
        float dd = q - x;
        dAcc += dd * dd;
        atomicAdd(&embsum[c * KCODES + kq], x);      // flat.T @ onehot
    }
    if (lane < 16) {
        int kq = sIdx[w * 16 + lane];
        atomicAdd(&counts[kq], 1.0f);
        outInd[n0 + lane] = (float)kq;
    }
    // wave-level diff reduction, one atomic per wave
    #pragma unroll
    for (int s = 1; s < 32; s <<= 1) dAcc += __shfl_xor(dAcc, s, 32);
    if (lane == 0) atomicAdd(wsDiff, dAcc);
}

// ---------------------------------------------------------------------------
// EMA cluster-size update + Laplace smoothing; also emits diff.
// ---------------------------------------------------------------------------
__global__ void vq_finalize_cs(const float* __restrict__ csIn,
                               const float* __restrict__ counts,
                               const float* __restrict__ wsDiff,
                               float* __restrict__ outCS,
                               float* __restrict__ outDiff,
                               float* __restrict__ csScaled)
{
    __shared__ float red[KCODES];
    int k = threadIdx.x;
    float ncs = csIn[k] * 0.99f + 0.01f * counts[k];
    red[k] = ncs;
    __syncthreads();
    for (int s = 256; s > 0; s >>= 1) {
        if (k < s) red[k] += red[k + s];
        __syncthreads();
    }
    float n = red[0];
    float cs = (ncs + 1e-5f) / (n + KCODES * 1e-5f) * n;
    outCS[k] = ncs;
    csScaled[k] = cs;
    if (k == 0) outDiff[0] = wsDiff[0] * (1.0f / 16777216.0f);  // mean over B*C*H*W
}

// ---------------------------------------------------------------------------
// EMA embedding_avg update + normalized embedding.
// ---------------------------------------------------------------------------
__global__ void vq_finalize_emb(const float* __restrict__ embAvgIn,
                                const float* __restrict__ embsum,
                                const float* __restrict__ csScaled,
                                float* __restrict__ outEA,
                                float* __restrict__ outEmb)
{
    int i = blockIdx.x * blockDim.x + threadIdx.x;   // 0..32767
    int k = i & (KCODES - 1);
    float nea = embAvgIn[i] * 0.99f + 0.01f * embsum[i];
    outEA[i] = nea;
    outEmb[i] = nea / csScaled[k];
}

// ---------------------------------------------------------------------------
extern "C" void kernel_launch(void* const* d_in, const int* in_sizes, int n_in,
                              void* d_out, int out_size, void* d_ws, size_t ws_size,
                              hipStream_t stream) {
    const float* input        = (const float*)d_in[0];   // 64*64*64*64
    const float* embedding    = (const float*)d_in[1];   // 64*512
    const float* cluster_size = (const float*)d_in[2];   // 512
    const float* embedding_avg= (const float*)d_in[3];   // 64*512

    float* out     = (float*)d_out;
    float* outQ    = out;                        // 16,777,216
    float* outDiff = outQ + 16777216;            // 1
    float* outInd  = outDiff + 1;                // 262,144
    float* outEmb  = outInd + 262144;            // 32,768
    float* outCS   = outEmb + 32768;             // 512
    float* outEA   = outCS + 512;                // 32,768

    float* ws       = (float*)d_ws;
    float* wsDiff   = ws;
    float* wsNorm   = ws + 1;
    float* wsCounts = ws + 513;
    float* wsEmbSum = ws + 1025;
    float* wsCs     = ws + 33793;

    vq_init_kernel<<<128, 256, 0, stream>>>(embedding, ws);
    vq_main_kernel<<<2048, 256, CDIM * KCODES * sizeof(float), stream>>>(
        input, embedding, wsNorm, wsCounts, wsEmbSum, wsDiff, outQ, outInd);
    vq_finalize_cs<<<1, 512, 0, stream>>>(cluster_size, wsCounts, wsDiff,
                                          outCS, outDiff, wsCs);
    vq_finalize_emb<<<128, 256, 0, stream>>>(embedding_avg, wsEmbSum, wsCs,
                                             outEA, outEmb);
}